// SimpleGNN_57380763074892
// MI455X (gfx1250) — compile-verified
//
#include <hip/hip_runtime.h>
#include <hip/hip_bf16.h>

typedef float v2f __attribute__((ext_vector_type(2)));
typedef float v8f __attribute__((ext_vector_type(8)));

#define N_NODES 50000
#define N_EDGES 800000
#define NUM_G   64
#define IN_DIM  128
#define H1      256
#define H2      128

__device__ __forceinline__ float lrelu(float x, float s) { return x > 0.0f ? x : s * x; }

__device__ __forceinline__ void atomAddF(float* p, float v) { unsafeAtomicAdd(p, v); }

// branch-free float atomic max via ordered-int trick (no CAS loop)
__device__ __forceinline__ void atomMaxF(float* p, float v) {
  if (v >= 0.0f) atomicMax((int*)p, __float_as_int(v));
  else           atomicMin((unsigned int*)p, __float_as_uint(v));
}

// ---------------- WMMA GEMM: C[M,NCOL] = A[M,K] @ B[K,NCOL] ----------------
// One wave computes a 16 x (16*NT) tile with V_WMMA_F32_16X16X4_F32.
template<int K, int NCOL, int NT>
__global__ void gemm_wmma_f32(const float* __restrict__ A, const float* __restrict__ B,
                              float* __restrict__ C, int M) {
  const int wave = (int)((blockIdx.x * blockDim.x + threadIdx.x) >> 5);
  const int lane = (int)(threadIdx.x & 31);
  constexpr int NGRP = NCOL / (16 * NT);
  const int mtile = wave / NGRP;
  const int ngrp  = wave % NGRP;
  if (mtile >= M / 16) return;              // wave-uniform exit, EXEC stays all-1 for WMMA

  const int row   = mtile * 16 + (lane & 15);
  const int khalf = (lane >> 4) * 2;        // lanes 0-15: K=k0,k0+1 ; lanes 16-31: K=k0+2,k0+3
  const int n0    = ngrp * 16 * NT;
  const int nIdx  = (lane & 15);

  v8f acc[NT];
#pragma unroll
  for (int t = 0; t < NT; ++t) acc[t] = v8f{};

  for (int k0 = 0; k0 < K; k0 += 4) {
    v2f a;
    a.x = A[row * K + k0 + khalf];
    a.y = A[row * K + k0 + khalf + 1];
#pragma unroll
    for (int t = 0; t < NT; ++t) {
      const int col = n0 + t * 16 + nIdx;
      v2f b;
      b.x = B[(k0 + khalf)     * NCOL + col];
      b.y = B[(k0 + khalf + 1) * NCOL + col];
      acc[t] = __builtin_amdgcn_wmma_f32_16x16x4_f32(false, a, false, b,
                                                     (short)0, acc[t], false, false);
    }
  }
  // C/D layout: VGPR v -> lanes 0-15: M=v, lanes 16-31: M=v+8; N = lane&15
  const int mOff = (lane >> 4) * 8;
#pragma unroll
  for (int t = 0; t < NT; ++t) {
    const int col = n0 + t * 16 + nIdx;
#pragma unroll
    for (int v = 0; v < 8; ++v) {
      C[(mtile * 16 + mOff + v) * NCOL + col] = acc[t][v];
    }
  }
}

// ---------------- GCN pieces ----------------
__global__ void k_deg_init(float* deg) {  // self-loop contributes 1
  int i = blockIdx.x * blockDim.x + threadIdx.x;
  if (i < N_NODES) deg[i] = 1.0f;
}
__global__ void k_deg_edges(const long long* __restrict__ dst, float* deg) {
  int e = blockIdx.x * blockDim.x + threadIdx.x;
  if (e < N_EDGES) atomAddF(&deg[(int)dst[e]], 1.0f);
}
__global__ void k_dinv(float* deg) {      // in place: dinv = rsqrt(deg), deg >= 1 always
  int i = blockIdx.x * blockDim.x + threadIdx.x;
  if (i < N_NODES) deg[i] = __frsqrt_rn(deg[i]);
}
__global__ void k_init_bias256(float* out1, const float* __restrict__ b1) {
  int i = blockIdx.x * blockDim.x + threadIdx.x;
  if (i < N_NODES * H1) out1[i] = b1[i & (H1 - 1)];
}
__global__ void k_gcn_scatter(const long long* __restrict__ src, const long long* __restrict__ dst,
                              const float* __restrict__ h, const float* __restrict__ dinv,
                              float* out1) {
  const int lane  = threadIdx.x & 31;
  const int wave0 = (int)((blockIdx.x * blockDim.x + threadIdx.x) >> 5);
  const int nwave = (int)((gridDim.x * blockDim.x) >> 5);
  for (int e = wave0; e < N_EDGES; e += nwave) {
    const int s = (int)src[e], d = (int)dst[e];
    const float norm = dinv[s] * dinv[d];
    const float* hs = h + (size_t)s * H1;
    float* od = out1 + (size_t)d * H1;
#pragma unroll
    for (int j = 0; j < H1 / 32; ++j) {
      const int c = lane + j * 32;
      atomAddF(&od[c], hs[c] * norm);
    }
  }
}
__global__ void k_gcn_finalize(float* x1, const float* __restrict__ h,
                               const float* __restrict__ dinv) {
  int idx = blockIdx.x * blockDim.x + threadIdx.x;
  if (idx < N_NODES * H1) {
    int i = idx / H1;
    float di = dinv[i];
    x1[idx] = lrelu(x1[idx] + h[idx] * di * di, 0.01f);   // self-loop + bias(already) + LeakyReLU
  }
}

// ---------------- GAT pieces ----------------
__global__ void k_att_scores(const float* __restrict__ g, const float* __restrict__ att_src,
                             const float* __restrict__ att_dst, float* a_s, float* a_d) {
  const int lane = threadIdx.x & 31;
  const int i    = (int)((blockIdx.x * blockDim.x + threadIdx.x) >> 5);
  if (i >= N_NODES) return;
  float ss = 0.0f, sd = 0.0f;
#pragma unroll
  for (int j = 0; j < H2 / 32; ++j) {
    const int c = lane + j * 32;
    const float v = g[(size_t)i * H2 + c];
    ss += v * att_src[c];
    sd += v * att_dst[c];
  }
#pragma unroll
  for (int off = 16; off > 0; off >>= 1) {
    ss += __shfl_down(ss, off, 32);
    sd += __shfl_down(sd, off, 32);
  }
  if (lane == 0) { a_s[i] = ss; a_d[i] = sd; }
}
__global__ void k_m_init(const float* __restrict__ a_s, const float* __restrict__ a_d, float* m) {
  int i = blockIdx.x * blockDim.x + threadIdx.x;
  if (i < N_NODES) m[i] = lrelu(a_s[i] + a_d[i], 0.2f);   // self-loop logit
}
__global__ void k_init_bias128(float* out2, const float* __restrict__ b2) {
  int i = blockIdx.x * blockDim.x + threadIdx.x;
  if (i < N_NODES * H2) out2[i] = b2[i & (H2 - 1)];
}
__global__ void k_edge_logit_max(const long long* __restrict__ src, const long long* __restrict__ dst,
                                 const float* __restrict__ a_s, const float* __restrict__ a_d,
                                 float* eval, float* m) {
  int e = blockIdx.x * blockDim.x + threadIdx.x;
  if (e < N_EDGES) {
    const int s = (int)src[e], d = (int)dst[e];
    const float v = lrelu(a_s[s] + a_d[d], 0.2f);
    eval[e] = v;
    atomMaxF(&m[d], v);
  }
}
__global__ void k_s_init(const float* __restrict__ a_s, const float* __restrict__ a_d,
                         const float* __restrict__ m, float* ssum) {
  int i = blockIdx.x * blockDim.x + threadIdx.x;
  if (i < N_NODES) ssum[i] = __expf(lrelu(a_s[i] + a_d[i], 0.2f) - m[i]);  // self-loop term
}
__global__ void k_edge_exp_sum(const long long* __restrict__ dst, const float* __restrict__ m,
                               float* eval, float* ssum) {
  int e = blockIdx.x * blockDim.x + threadIdx.x;
  if (e < N_EDGES) {
    const int d = (int)dst[e];
    const float ex = __expf(eval[e] - m[d]);
    eval[e] = ex;
    atomAddF(&ssum[d], ex);
  }
}
__global__ void k_gat_scatter(const long long* __restrict__ src, const long long* __restrict__ dst,
                              const float* __restrict__ g, const float* __restrict__ eval,
                              const float* __restrict__ ssum, float* out2) {
  const int lane  = threadIdx.x & 31;
  const int wave0 = (int)((blockIdx.x * blockDim.x + threadIdx.x) >> 5);
  const int nwave = (int)((gridDim.x * blockDim.x) >> 5);
  for (int e = wave0; e < N_EDGES; e += nwave) {
    const int s = (int)src[e], d = (int)dst[e];
    const float alpha = eval[e] / ssum[d];
    const float* gs = g + (size_t)s * H2;
    float* od = out2 + (size_t)d * H2;
#pragma unroll
    for (int j = 0; j < H2 / 32; ++j) {
      const int c = lane + j * 32;
      atomAddF(&od[c], gs[c] * alpha);
    }
  }
}
__global__ void k_gat_finalize(float* x2, const float* __restrict__ g,
                               const float* __restrict__ a_s, const float* __restrict__ a_d,
                               const float* __restrict__ m, const float* __restrict__ ssum) {
  int idx = blockIdx.x * blockDim.x + threadIdx.x;
  if (idx < N_NODES * H2) {
    const int i = idx / H2;
    const float alpha_self = __expf(lrelu(a_s[i] + a_d[i], 0.2f) - m[i]) / ssum[i];
    x2[idx] = lrelu(x2[idx] + g[idx] * alpha_self, 0.01f);
  }
}

// ---------------- pooling + head ----------------
__global__ void k_pool_init(float* pooled, float* cnt) {
  int i = blockIdx.x * blockDim.x + threadIdx.x;
  if (i < NUM_G * H2) pooled[i] = 0.0f;
  if (i < NUM_G) cnt[i] = 0.0f;
}
__global__ void k_pool_scatter(const long long* __restrict__ batch, const float* __restrict__ x2,
                               float* pooled, float* cnt) {
  const int lane  = threadIdx.x & 31;
  const int wave0 = (int)((blockIdx.x * blockDim.x + threadIdx.x) >> 5);
  const int nwave = (int)((gridDim.x * blockDim.x) >> 5);
  for (int i = wave0; i < N_NODES; i += nwave) {
    const int b = (int)batch[i];
    const float* xi = x2 + (size_t)i * H2;
    float* pb = pooled + (size_t)b * H2;
#pragma unroll
    for (int j = 0; j < H2 / 32; ++j) {
      const int c = lane + j * 32;
      atomAddF(&pb[c], xi[c]);
    }
    if (lane == 0) atomAddF(&cnt[b], 1.0f);
  }
}
__global__ void k_head(const float* __restrict__ pooled, const float* __restrict__ cnt,
                       const float* __restrict__ Wl, const float* __restrict__ bl, float* out) {
  const int lane = threadIdx.x & 31;
  const int gidx = (int)((blockIdx.x * blockDim.x + threadIdx.x) >> 5);
  if (gidx >= NUM_G) return;
  const float inv = 1.0f / fmaxf(cnt[gidx], 1.0f);
  float acc = 0.0f;
#pragma unroll
  for (int j = 0; j < H2 / 32; ++j) {
    const int c = lane + j * 32;
    acc += pooled[(size_t)gidx * H2 + c] * inv * Wl[c];
  }
#pragma unroll
  for (int off = 16; off > 0; off >>= 1) acc += __shfl_down(acc, off, 32);
  if (lane == 0) out[gidx] = acc + bl[0];
}

extern "C" void kernel_launch(void* const* d_in, const int* in_sizes, int n_in,
                              void* d_out, int out_size, void* d_ws, size_t ws_size,
                              hipStream_t stream) {
  const float*     x    = (const float*)d_in[0];
  const long long* ei   = (const long long*)d_in[1];   // int64 [2, E]
  const long long* batch= (const long long*)d_in[2];   // int64 [N]
  const float*     W1   = (const float*)d_in[4];
  const float*     b1   = (const float*)d_in[5];
  const float*     W2   = (const float*)d_in[6];
  const float*     atts = (const float*)d_in[7];
  const float*     attd = (const float*)d_in[8];
  const float*     b2   = (const float*)d_in[9];
  const float*     Wl   = (const float*)d_in[10];
  const float*     bl   = (const float*)d_in[11];
  float* out = (float*)d_out;

  const long long* src = ei;
  const long long* dst = ei + N_EDGES;

  // workspace carve-up (floats)
  float* p = (float*)d_ws;
  float* dinv   = p;               p += N_NODES;
  float* h      = p;               p += (size_t)N_NODES * H1;
  float* x1     = p;               p += (size_t)N_NODES * H1;   // also out1 (in-place)
  float* g      = p;               p += (size_t)N_NODES * H2;
  float* a_s    = p;               p += N_NODES;
  float* a_d    = p;               p += N_NODES;
  float* m      = p;               p += N_NODES;
  float* ssum   = p;               p += N_NODES;
  float* eval   = p;               p += N_EDGES;
  float* out2   = p;               p += (size_t)N_NODES * H2;   // also x2 (in-place)
  float* pooled = p;               p += (size_t)NUM_G * H2;
  float* cnt    = p;               p += NUM_G;

  const int T = 256;
  const int gN    = (N_NODES + T - 1) / T;
  const int gE    = (N_EDGES + T - 1) / T;
  const int gN256 = (N_NODES * H1 + T - 1) / T;
  const int gN128 = (N_NODES * H2 + T - 1) / T;
  const int gWaveN = ((N_NODES * 32) + T - 1) / T;   // one wave per node
  const int gScat  = 4096;                           // grid-stride wave-per-edge kernels

  // ---- GCN ----
  k_deg_init   <<<gN, T, 0, stream>>>(dinv);
  k_deg_edges  <<<gE, T, 0, stream>>>(dst, dinv);
  k_dinv       <<<gN, T, 0, stream>>>(dinv);
  k_init_bias256<<<gN256, T, 0, stream>>>(x1, b1);
  {
    constexpr int NT = 4, NGRP = H1 / (16 * NT);     // 4 groups of 64 cols
    const int waves = (N_NODES / 16) * NGRP;
    const int blocks = (waves * 32 + T - 1) / T;
    gemm_wmma_f32<IN_DIM, H1, NT><<<blocks, T, 0, stream>>>(x, W1, h, N_NODES);
  }
  k_gcn_scatter <<<gScat, T, 0, stream>>>(src, dst, h, dinv, x1);
  k_gcn_finalize<<<gN256, T, 0, stream>>>(x1, h, dinv);

  // ---- GAT ----
  {
    constexpr int NT = 4, NGRP = H2 / (16 * NT);     // 2 groups of 64 cols
    const int waves = (N_NODES / 16) * NGRP;
    const int blocks = (waves * 32 + T - 1) / T;
    gemm_wmma_f32<H1, H2, NT><<<blocks, T, 0, stream>>>(x1, W2, g, N_NODES);
  }
  k_att_scores   <<<gWaveN, T, 0, stream>>>(g, atts, attd, a_s, a_d);
  k_m_init       <<<gN, T, 0, stream>>>(a_s, a_d, m);
  k_init_bias128 <<<gN128, T, 0, stream>>>(out2, b2);
  k_edge_logit_max<<<gE, T, 0, stream>>>(src, dst, a_s, a_d, eval, m);
  k_s_init       <<<gN, T, 0, stream>>>(a_s, a_d, m, ssum);
  k_edge_exp_sum <<<gE, T, 0, stream>>>(dst, m, eval, ssum);
  k_gat_scatter  <<<gScat, T, 0, stream>>>(src, dst, g, eval, ssum, out2);
  k_gat_finalize <<<gN128, T, 0, stream>>>(out2, g, a_s, a_d, m, ssum);

  // ---- pool + head ----
  k_pool_init   <<<(NUM_G * H2 + T - 1) / T, T, 0, stream>>>(pooled, cnt);
  k_pool_scatter<<<gScat, T, 0, stream>>>(batch, out2, pooled, cnt);
  k_head        <<<(NUM_G * 32 + T - 1) / T, T, 0, stream>>>(pooled, cnt, Wl, bl, out);

  (void)in_sizes; (void)n_in; (void)out_size; (void)ws_size;
}